// PtransE_c_42992622633013
// MI455X (gfx1250) — compile-verified
//
#include <hip/hip_runtime.h>
#include <stdint.h>

#define DIM      64
#define WAVE     32
#define GAMMA_F  1.0f
#define TILE     16          // triples per k_loss block
#define LDSTRIDE 68          // LDS row stride (floats): 68 mod 64 = 4 -> conflict-free

typedef uint32_t u32x4 __attribute__((ext_vector_type(4)));
typedef int32_t  i32x8 __attribute__((ext_vector_type(8)));
typedef int32_t  i32x4 __attribute__((ext_vector_type(4)));
typedef float    v2f   __attribute__((ext_vector_type(2)));
typedef float    v8f   __attribute__((ext_vector_type(8)));

// Native fp32 global atomic add (global_atomic_add_f32, no CAS loop).
__device__ __forceinline__ void atomAddF(float* p, float v) {
    unsafeAtomicAdd(p, v);
}

// ---------------------------------------------------------------------------
// Kernel 0: zero paths_factor scratch and the scalar output.
// ---------------------------------------------------------------------------
__global__ void k_zero(float* __restrict__ paths, int n, float* __restrict__ out) {
    int i = blockIdx.x * blockDim.x + threadIdx.x;
    if (i < n) paths[i] = 0.0f;
    if (i == 0) out[0] = 0.0f;
}

// ---------------------------------------------------------------------------
// Stage RD floats of relation_emb into LDS.
// Preferred: one Tensor-Data-Mover DMA descriptor issued by wave 0
// (TENSOR_LOAD_TO_LDS, tracked by TENSORcnt).
// Fallback: per-thread async global->LDS b128 copies (ASYNCcnt).
// ---------------------------------------------------------------------------
__device__ __forceinline__ void stage_relation_lds(float* s_rel,
                                                   const float* __restrict__ relation_emb,
                                                   int RD) {
#if __has_builtin(__builtin_amdgcn_tensor_load_to_lds)
    if (threadIdx.x < WAVE) {          // one wave issues the DMA (EXEC ignored by TDM)
        uint32_t lds_addr = (uint32_t)(uintptr_t)(&s_rel[0]);   // wave-relative LDS bytes
        uint64_t gaddr    = (uint64_t)(uintptr_t)relation_emb;

        // ---- D# group 0 (128b): count=1 | lds_addr | global_addr[56:0] | type=2
        u32x4 g0;
        g0.x = 1u;                                   // count=1 (valid descriptor)
        g0.y = lds_addr;                             // bits 63:32 lds_addr
        g0.z = (uint32_t)(gaddr & 0xFFFFFFFFu);      // global_addr[31:0]
        g0.w = ((uint32_t)(gaddr >> 32) & 0x01FFFFFFu) | (2u << 30);  // addr[56:32]|type=2

        // ---- D# group 1 (256b): 1-D tile of RD elements, data_size=4B (code 2)
        uint32_t dim0 = (uint32_t)RD;                // 64000 < 2^16
        i32x8 g1;
        g1[0] = (int32_t)(2u << 16);                 // workgroup_mask=0, data_size=2
        g1[1] = (int32_t)((dim0 & 0xFFFFu) << 16);   // tensor_dim0[15:0] in bits 63:48
        g1[2] = (int32_t)(((dim0 >> 16) & 0xFFFFu)   // tensor_dim0[31:16]
                 | (1u << 16));                      // tensor_dim1 = 1
        g1[3] = (int32_t)((dim0 & 0xFFFFu) << 16);   // tensor_dim1 hi=0, tile_dim0=RD
        g1[4] = (int32_t)1;                          // tile_dim1 = 1, tile_dim2 = 0
        g1[5] = (int32_t)dim0;                       // tensor_dim0_stride[31:0]
        g1[6] = 0;                                   // stride hi, tensor_dim1_stride lo
        g1[7] = 0;

        i32x4 z4 = {0, 0, 0, 0};                     // groups 2/3 unused (dims >= 2 off)
        i32x8 z8 = {0, 0, 0, 0, 0, 0, 0, 0};         // extra operand (clang-23 6-arg form)
        __builtin_amdgcn_tensor_load_to_lds(g0, g1, z4, z4, z8, 0);
        __builtin_amdgcn_s_wait_tensorcnt(0);
    }
    __syncthreads();
#else
    uint32_t ldsbase = (uint32_t)(uintptr_t)(&s_rel[0]);
    int nchunk = RD >> 2;                            // float4 chunks
    for (int i = threadIdx.x; i < nchunk; i += blockDim.x) {
        uint64_t src = (uint64_t)(uintptr_t)(relation_emb + (i << 2));
        uint32_t dst = ldsbase + ((uint32_t)i << 4);
        asm volatile("global_load_async_to_lds_b128 %0, %1, off"
                     :: "v"(dst), "v"(src) : "memory");
    }
    asm volatile("s_wait_asynccnt 0" ::: "memory");
    __syncthreads();
#endif
}

// ---------------------------------------------------------------------------
// Kernel 1: segment-sum of weighted relation rows out of LDS.
// Launch: 256 threads/block, dynamic LDS = RD*4 bytes.
// ---------------------------------------------------------------------------
__global__ void k_segsum(const int*   __restrict__ path_rels,
                         const float* __restrict__ step_weight,
                         const int*   __restrict__ step_triple,
                         const float* __restrict__ relation_emb,
                         float*       __restrict__ paths,
                         int T, int RD) {
    extern __shared__ __align__(16) float s_rel[];
    stage_relation_lds(s_rel, relation_emb, RD);

    const int waves_per_block = blockDim.x / WAVE;
    const int wave        = blockIdx.x * waves_per_block + (int)(threadIdx.x / WAVE);
    const int total_waves = gridDim.x * waves_per_block;
    const int span        = (T + total_waves - 1) / total_waves;
    const int start       = wave * span;
    const int end         = (start + span < T) ? (start + span) : T;
    const int lane        = threadIdx.x & (WAVE - 1);

    if (start >= end) return;

    float ax = 0.0f, ay = 0.0f;
    int   cur = step_triple[start];

    for (int t = start; t < end; ++t) {
        int trip = step_triple[t];
        if (trip != cur) {
            float* p = paths + (size_t)cur * DIM + 2 * lane;
            atomAddF(p,     ax);
            atomAddF(p + 1, ay);
            ax = 0.0f; ay = 0.0f;
            cur = trip;
        }
        float w = step_weight[t];
        int   r = path_rels[t];
        float2 v = *(const float2*)(s_rel + (size_t)r * DIM + 2 * lane);  // ds_load_b64
        ax = fmaf(w, v.x, ax);
        ay = fmaf(w, v.y, ay);
    }
    float* p = paths + (size_t)cur * DIM + 2 * lane;
    atomAddF(p,     ax);
    atomAddF(p + 1, ay);
}

// ---------------------------------------------------------------------------
// Kernel 2: per-triple distances + norms + hinge loss.
// Block = 512 threads = 16 waves = 16 triples. Each wave gathers its triple's
// rows (float2/lane, coalesced 256B) and writes its 64-dim distance vector to
// an LDS tile. The rowwise sum-of-squares is the diagonal of X*X^T, computed
// with 16 accumulating V_WMMA_F32_16X16X4_F32 ops (A == B register-identical
// since the 16x4 A layout and 4x16 B layout are mutual transposes).
// ---------------------------------------------------------------------------
__global__ void k_loss(const int* __restrict__ ph, const int* __restrict__ pr,
                       const int* __restrict__ pt, const int* __restrict__ nh,
                       const int* __restrict__ nr, const int* __restrict__ nt,
                       const float* __restrict__ entity_emb,
                       const float* __restrict__ relation_emb,
                       const float* __restrict__ type_emb,
                       const float* __restrict__ paths,
                       float* __restrict__ out, int Bn) {
    __shared__ float s_pos[TILE][LDSTRIDE];
    __shared__ float s_neg[TILE][LDSTRIDE];
    __shared__ float s_pn[TILE];
    __shared__ float s_nn[TILE];

    const int wave = (int)(threadIdx.x / WAVE);       // 0..15 = triple within tile
    const int lane = threadIdx.x & (WAVE - 1);
    const int base = blockIdx.x * TILE;
    const int valid = (Bn - base < TILE) ? (Bn - base) : TILE;
    const int b  = base + wave;
    const int bc = (b < Bn) ? b : (Bn - 1);           // clamp; masked at the end
    const int d0 = 2 * lane;

    int iph = ph[bc], ipt = pt[bc], ipr = pr[bc];
    int inh = nh[bc], int_ = nt[bc], inr = nr[bc];

    float2 e_ph = *(const float2*)(entity_emb + (size_t)iph * DIM + d0);
    float2 t_ph = *(const float2*)(type_emb   + (size_t)iph * DIM + d0);
    float2 e_pt = *(const float2*)(entity_emb + (size_t)ipt * DIM + d0);
    float2 t_pt = *(const float2*)(type_emb   + (size_t)ipt * DIM + d0);
    float2 r_p  = *(const float2*)(relation_emb + (size_t)ipr * DIM + d0);
    float2 pf   = *(const float2*)(paths      + (size_t)bc  * DIM + d0);

    float px = e_ph.x * t_ph.x + r_p.x + pf.x - e_pt.x * t_pt.x;
    float py = e_ph.y * t_ph.y + r_p.y + pf.y - e_pt.y * t_pt.y;

    float2 e_nh = *(const float2*)(entity_emb + (size_t)inh * DIM + d0);
    float2 t_nh = *(const float2*)(type_emb   + (size_t)inh * DIM + d0);
    float2 e_nt = *(const float2*)(entity_emb + (size_t)int_ * DIM + d0);
    float2 t_nt = *(const float2*)(type_emb   + (size_t)int_ * DIM + d0);
    float2 r_n  = *(const float2*)(relation_emb + (size_t)inr * DIM + d0);

    float nx = e_nh.x * t_nh.x + r_n.x - e_nt.x * t_nt.x;
    float ny = e_nh.y * t_nh.y + r_n.y - e_nt.y * t_nt.y;

#if __has_builtin(__builtin_amdgcn_wmma_f32_16x16x4_f32)
    s_pos[wave][d0]     = px;
    s_pos[wave][d0 + 1] = py;
    s_neg[wave][d0]     = nx;
    s_neg[wave][d0 + 1] = ny;
    __syncthreads();

    if (wave < 2) {                       // wave 0: pos tile, wave 1: neg tile
        const float* Xs = (wave == 0) ? &s_pos[0][0] : &s_neg[0][0];
        const int m = lane & 15;          // matrix row
        const int h = lane >> 4;          // K half-select
        v8f c = {0.f, 0.f, 0.f, 0.f, 0.f, 0.f, 0.f, 0.f};
        #pragma unroll
        for (int kk = 0; kk < 16; ++kk) {
            // A(m,k): vgpr=k&1, lane=m+16*(k>>1)  ->  lane loads (k, k+1) pair
            float2 af = *(const float2*)(Xs + (size_t)m * LDSTRIDE + 4 * kk + 2 * h);
            v2f a; a.x = af.x; a.y = af.y;
            // Gram accumulate: D = A*A^T + C   (A/B layouts are mutual transposes)
            c = __builtin_amdgcn_wmma_f32_16x16x4_f32(
                    false, a, false, a, (short)0, c, false, false);
        }
        // diagonal extraction: diag(i<8) at vgpr i / lane i,
        //                      diag(i>=8) at vgpr i-8 / lane i+16
        float d = 0.0f;
        #pragma unroll
        for (int j = 0; j < 8; ++j) {
            float t  = c[j];
            float lo = __shfl(t, j,      WAVE);
            float hi = __shfl(t, j + 24, WAVE);
            d = (lane == j)     ? lo : d;
            d = (lane == j + 8) ? hi : d;
        }
        float nrm = sqrtf(d);
        if (lane < TILE) {
            if (wave == 0) s_pn[lane] = nrm; else s_nn[lane] = nrm;
        }
    }
    __syncthreads();
#else
    float ssp = px * px + py * py;
    float ssn = nx * nx + ny * ny;
    #pragma unroll
    for (int off = 16; off >= 1; off >>= 1) {
        ssp += __shfl_xor(ssp, off, WAVE);
        ssn += __shfl_xor(ssn, off, WAVE);
    }
    if (lane == 0) {
        s_pn[wave] = sqrtf(ssp);
        s_nn[wave] = sqrtf(ssn);
    }
    __syncthreads();
#endif

    if (wave == 0) {                      // all 32 lanes active; mask beyond `valid`
        float contrib = 0.0f;
        if (lane < valid) {
            float pn = s_pn[lane & (TILE - 1)];
            float nn = s_nn[lane & (TILE - 1)];
            float hinge = GAMMA_F + pn - nn;
            contrib = (hinge > 0.0f ? hinge : 0.0f) + 0.001f * (pn + nn);
        }
        #pragma unroll
        for (int off = 16; off >= 1; off >>= 1)
            contrib += __shfl_xor(contrib, off, WAVE);
        if (lane == 0) atomAddF(out, contrib);
    }
}

// ---------------------------------------------------------------------------
// Host-side launch (graph-capture safe: only kernel launches on `stream`).
// ---------------------------------------------------------------------------
extern "C" void kernel_launch(void* const* d_in, const int* in_sizes, int n_in,
                              void* d_out, int out_size, void* d_ws, size_t ws_size,
                              hipStream_t stream) {
    const int*   pos_head     = (const int*)  d_in[0];
    const int*   pos_relation = (const int*)  d_in[1];
    const int*   pos_tail     = (const int*)  d_in[2];
    const int*   neg_head     = (const int*)  d_in[3];
    const int*   neg_relation = (const int*)  d_in[4];
    const int*   neg_tail     = (const int*)  d_in[5];
    const int*   path_rels    = (const int*)  d_in[6];
    const float* step_weight  = (const float*)d_in[7];
    const int*   step_triple  = (const int*)  d_in[8];
    const float* entity_emb   = (const float*)d_in[9];
    const float* relation_emb = (const float*)d_in[10];
    const float* type_emb     = (const float*)d_in[11];

    const int Bn = in_sizes[0];    // 4096 triples
    const int T  = in_sizes[6];    // 98304 path steps
    const int RD = in_sizes[10];   // R*D = 64000 floats (256 KB)

    float* paths = (float*)d_ws;   // B*DIM floats of scratch
    float* out   = (float*)d_out;

    // allow 256 KB of dynamic LDS (gfx1250 WGP has 320 KB)
    size_t ldsBytes = (size_t)RD * sizeof(float);
    (void)hipFuncSetAttribute((const void*)k_segsum,
                              hipFuncAttributeMaxDynamicSharedMemorySize,
                              (int)ldsBytes);

    const int nPaths = Bn * DIM;
    k_zero<<<(nPaths + 255) / 256, 256, 0, stream>>>(paths, nPaths, out);

    const int segBlocks = 32;      // 256 waves -> 384 steps per wave
    k_segsum<<<segBlocks, 256, ldsBytes, stream>>>(
        path_rels, step_weight, step_triple, relation_emb, paths, T, RD);

    const int lossBlocks = (Bn + TILE - 1) / TILE;   // 16 triples per 512-thread block
    k_loss<<<lossBlocks, 512, 0, stream>>>(
        pos_head, pos_relation, pos_tail, neg_head, neg_relation, neg_tail,
        entity_emb, relation_emb, type_emb, paths, out, Bn);
}